// Model_69767448756494
// MI455X (gfx1250) — compile-verified
//
#include <hip/hip_runtime.h>

#define VOCAB_N 128000
#define BATCH_N 256
#define CHUNK   2048
#define NCHUNK  ((VOCAB_N + CHUNK - 1) / CHUNK)   /* 63, last chunk = 1024 */
#define NWAVE   8
#define NEG_INF (-__builtin_huge_valf())

typedef float v2f __attribute__((ext_vector_type(2)));
typedef float v8f __attribute__((ext_vector_type(8)));

__device__ __forceinline__ float wave_sum_f(float v) {
#pragma unroll
  for (int o = 16; o > 0; o >>= 1) v += __shfl_xor(v, o, 32);
  return v;
}
__device__ __forceinline__ int wave_sum_i(int v) {
#pragma unroll
  for (int o = 16; o > 0; o >>= 1) v += __shfl_xor(v, o, 32);
  return v;
}

// One block (256 threads = 8 wave32) per row. Computes cutoff index J[row].
__global__ __launch_bounds__(256) void topkp_cutoff_kernel(
    const float* __restrict__ sorted_value,
    const float* __restrict__ p_arr,
    const int*   __restrict__ k_arr,
    int*         __restrict__ Jout) {
  const int r    = blockIdx.x;
  const int tid  = threadIdx.x;
  const int lane = tid & 31;
  const int w    = tid >> 5;
  const float* __restrict__ row = sorted_value + (size_t)r * VOCAB_N;

  __shared__ float s_chunk[NCHUNK];
  __shared__ float s_wtot[NWAVE];
  __shared__ int   s_mcnt;
  __shared__ int   s_pcnt;
  if (tid == 0) { s_mcnt = 0; s_pcnt = 0; }
  __syncthreads();

  const int  kv     = k_arr[r];
  const bool validk = (kv >= 1) && (kv <= VOCAB_N);
  int ti = VOCAB_N - kv;
  ti = ti < 0 ? 0 : (ti > VOCAB_N - 1 ? VOCAB_N - 1 : ti);
  const float thr  = validk ? row[ti] : NEG_INF;   // -inf => top-k never masks
  const float vmax = row[VOCAB_N - 1];             // row max (ascending sorted)

  // ---- Pass A: per-chunk sums of masked exp(v - vmax); count top-k-masked elems.
  // Chunks are striped over waves so each chunk sum needs no cross-wave barrier.
  int mcnt = 0;
  for (int c = w; c < NCHUNK; c += NWAVE) {
    const int base = c * CHUNK;
    const int rem  = (VOCAB_N - base < CHUNK) ? (VOCAB_N - base) : CHUNK;
    float acc = 0.f;
#pragma unroll
    for (int i = 0; i < CHUNK / 128; ++i) {     // 16 float4 iters per lane
      const int off = (i * 32 + lane) * 4;
      if (off < rem) {                          // rem is a multiple of 1024
        const float4 v = *(const float4*)(row + base + off);
        acc += (v.x < thr) ? 0.f : __expf(v.x - vmax); mcnt += (v.x < thr);
        acc += (v.y < thr) ? 0.f : __expf(v.y - vmax); mcnt += (v.y < thr);
        acc += (v.z < thr) ? 0.f : __expf(v.z - vmax); mcnt += (v.z < thr);
        acc += (v.w < thr) ? 0.f : __expf(v.w - vmax); mcnt += (v.w < thr);
      }
    }
    const float csum = wave_sum_f(acc);
    if (lane == 0) s_chunk[c] = csum;
  }
  mcnt = wave_sum_i(mcnt);
  if (lane == 0) atomicAdd(&s_mcnt, mcnt);
  __syncthreads();

  // ---- Locate crossing chunk cc (redundantly, identically, in every thread).
  float S = 0.f;
  for (int c = 0; c < NCHUNK; ++c) S += s_chunk[c];
  const float pv  = p_arr[r];
  const float tgt = (1.0f - pv) * S;            // cumsum_exp <= (1-p)*S

  float run = 0.f, R = 0.f;
  int cc = NCHUNK - 1;
  bool found = false;
  for (int c = 0; c < NCHUNK; ++c) {
    const float nr = run + s_chunk[c];
    if (!found && nr > tgt) { cc = c; R = run; found = true; }
    run = nr;
  }
  if (!found) { cc = NCHUNK - 1; R = run - s_chunk[NCHUNK - 1]; }

  // ---- Fine scan of the crossing chunk via V_WMMA_F32_16X16X4_F32.
  // Wave w scans its 256-element subblock laid out as M[k][n] = e[16n + k].
  // Phase 1: D1 = Lones(16x16) x M   -> inclusive scan within each 16-group
  // Phase 2: D  = D1 + (1 * t^T) x Ustrict -> add exclusive group offsets
  const int sb  = cc * CHUNK + w * 256;
  const int m16 = lane & 15;
  const int hl  = lane >> 4;                    // lane half: 0 or 1
  v8f acc8 = {0.f, 0.f, 0.f, 0.f, 0.f, 0.f, 0.f, 0.f};
#pragma unroll
  for (int s = 0; s < 4; ++s) {                 // K-slices of 4
    const int kb = 4 * s + 2 * hl;
    v2f a, b;
    // A slice of lower-triangular ones: A[m][k] = (k <= m)
    a.x = (kb     <= m16) ? 1.f : 0.f;
    a.y = (kb + 1 <= m16) ? 1.f : 0.f;
    // B slice = data: B[k][n] = e[16n + k]; this lane supplies e[g0], e[g0+1]
    const int g0 = sb + 16 * m16 + kb;
    float v0 = 0.f, v1 = 0.f;
    if (g0 + 1 < VOCAB_N) {
      const float2 vv = *(const float2*)(row + g0);
      v0 = vv.x; v1 = vv.y;
    } else if (g0 < VOCAB_N) {
      v0 = row[g0];
    }
    b.x = (g0     < VOCAB_N) ? ((v0 < thr) ? 0.f : __expf(v0 - vmax)) : 0.f;
    b.y = (g0 + 1 < VOCAB_N) ? ((v1 < thr) ? 0.f : __expf(v1 - vmax)) : 0.f;
    acc8 = __builtin_amdgcn_wmma_f32_16x16x4_f32(
        false, a, false, b, (short)0, acc8, false, false);
  }
  // Group totals t[n] = D1[15][n] live in VGPR7 of lanes 16+n.
  const float t7 = acc8[7];
#pragma unroll
  for (int s = 0; s < 4; ++s) {
    const int kb = 4 * s + 2 * hl;
    v2f a, b;
    a.x = __shfl(t7, 16 + kb,     32);          // A[m][k] = t[k] (rows broadcast)
    a.y = __shfl(t7, 16 + kb + 1, 32);
    b.x = (kb     < m16) ? 1.f : 0.f;           // B[k][n] = (k < n) strict upper
    b.y = (kb + 1 < m16) ? 1.f : 0.f;
    acc8 = __builtin_amdgcn_wmma_f32_16x16x4_f32(
        false, a, false, b, (short)0, acc8, false, false);
  }
  // acc8[j] on lane L = inclusive scan of subblock element 16*(L&15) + 8*(L>>4) + j.
  const float wtot = __shfl(acc8[7], 31, 32);   // scan of element 255 = wave total
  if (lane == 0) s_wtot[w] = wtot;
  __syncthreads();

  float wbase = R;
  for (int i = 0; i < w; ++i) wbase += s_wtot[i];
  const float tw = tgt - wbase;
  int cnt = 0;
#pragma unroll
  for (int j = 0; j < 8; ++j) cnt += (acc8[j] <= tw) ? 1 : 0;
  cnt = wave_sum_i(cnt);
  if (lane == 0) atomicAdd(&s_pcnt, cnt);
  __syncthreads();

  if (tid == 0) {
    const int c_all = cc * CHUNK + s_pcnt;                    // top-p prefix len
    const int topp  = (c_all < VOCAB_N - 1) ? c_all : (VOCAB_N - 1);
    const int m     = validk ? s_mcnt : 0;                    // top-k prefix len
    Jout[r] = (m > topp) ? m : topp;
  }
}

// out[r, pos] = (si[r,pos] < J[r]) ? -inf : sv[r, si[r,pos]]
// sorted_value (131 MB) is L2-resident on MI455X (192 MB L2) -> gather hits L2.
__global__ __launch_bounds__(256) void topkp_gather_kernel(
    const float* __restrict__ sorted_value,
    const int*   __restrict__ sorted_indices,
    const int*   __restrict__ Jin,
    float*       __restrict__ out) {
  const int r   = blockIdx.y;
  const int pos = (blockIdx.x * blockDim.x + threadIdx.x) * 4;
  const size_t rb = (size_t)r * VOCAB_N;
  const int J = Jin[r];
  const int4 idx = *(const int4*)(sorted_indices + rb + pos);
  const float* __restrict__ row = sorted_value + rb;
  float4 o;
  o.x = (idx.x < J) ? NEG_INF : row[idx.x];
  o.y = (idx.y < J) ? NEG_INF : row[idx.y];
  o.z = (idx.z < J) ? NEG_INF : row[idx.z];
  o.w = (idx.w < J) ? NEG_INF : row[idx.w];
  *(float4*)(out + rb + pos) = o;
}

extern "C" void kernel_launch(void* const* d_in, const int* in_sizes, int n_in,
                              void* d_out, int out_size, void* d_ws, size_t ws_size,
                              hipStream_t stream) {
  (void)in_sizes; (void)n_in; (void)out_size; (void)ws_size;
  const float* sv = (const float*)d_in[0];
  const int*   si = (const int*)d_in[1];
  const float* p  = (const float*)d_in[2];
  const int*   k  = (const int*)d_in[3];
  float* out = (float*)d_out;
  int*   J   = (int*)d_ws;   // 256 ints of scratch

  topkp_cutoff_kernel<<<dim3(BATCH_N), dim3(256), 0, stream>>>(sv, p, k, J);
  topkp_gather_kernel<<<dim3(VOCAB_N / 1024, BATCH_N), dim3(256), 0, stream>>>(
      sv, si, J, out);
}